// HybridDeepQ_21938692948017
// MI455X (gfx1250) — compile-verified
//
#include <hip/hip_runtime.h>
#include <hip/hip_bf16.h>

// ---------------------------------------------------------------------------
// HybridDeepQ for MI455X (gfx1250, wave32).
//  Phase 1: per-lane 7-qubit statevector sim, 128 f32 amplitudes in VGPRs,
//           fully unrolled gates (compile-time indices -> pure register ops).
//  Phase 2: 3-layer MLP via V_WMMA_F32_16X16X4_F32 (f32 precision, K=4/issue),
//           A-panel and inter-layer activations staged through padded LDS.
// ---------------------------------------------------------------------------

typedef __attribute__((ext_vector_type(2))) float v2f;
typedef __attribute__((ext_vector_type(8))) float v8f;

#define NQ 7
#define NSTATE 128

// RY(theta) on the qubit whose amplitude-index bit is BIT (qubit q -> 64>>q).
template<int BIT>
__device__ __forceinline__ void ry_gate(float* s, float c, float sn) {
#pragma unroll
  for (int i = 0; i < NSTATE; ++i) {
    if ((i & BIT) == 0) {
      float a0 = s[i];
      float a1 = s[i | BIT];
      s[i]       = c * a0 - sn * a1;
      s[i | BIT] = sn * a0 + c * a1;
    }
  }
}

// CNOT: swap target pair where control bit is set. Pure register renaming.
template<int CBIT, int TBIT>
__device__ __forceinline__ void cnot_gate(float* s) {
#pragma unroll
  for (int i = 0; i < NSTATE; ++i) {
    if ((i & CBIT) != 0 && (i & TBIT) == 0) {
      float t = s[i];
      s[i] = s[i | TBIT];
      s[i | TBIT] = t;
    }
  }
}

__device__ __forceinline__ void ry_all(float* s, const float* c, const float* sn) {
  ry_gate<64>(s, c[0], sn[0]);
  ry_gate<32>(s, c[1], sn[1]);
  ry_gate<16>(s, c[2], sn[2]);
  ry_gate< 8>(s, c[3], sn[3]);
  ry_gate< 4>(s, c[4], sn[4]);
  ry_gate< 2>(s, c[5], sn[5]);
  ry_gate< 1>(s, c[6], sn[6]);
}

__device__ __forceinline__ void cnot_ring(float* s) {
  cnot_gate<64, 32>(s);   // (0,1)
  cnot_gate<32, 16>(s);   // (1,2)
  cnot_gate<16,  8>(s);   // (2,3)
  cnot_gate< 8,  4>(s);   // (3,4)
  cnot_gate< 4,  2>(s);   // (4,5)
  cnot_gate< 2,  1>(s);   // (5,6)
  cnot_gate< 1, 64>(s);   // (6,0)
}

__global__ __launch_bounds__(128)
void hybrid_deepq_kernel(const float* __restrict__ x_q,
                         const float* __restrict__ x_c,
                         const float* __restrict__ q_params,
                         const float* __restrict__ w1, const float* __restrict__ b1,
                         const float* __restrict__ w2, const float* __restrict__ b2,
                         const float* __restrict__ w3, const float* __restrict__ b3,
                         float* __restrict__ out) {
  // A-panel: 128 rows x 20 cols (17 features zero-padded to 20 = 5 K-chunks).
  // Stride 20 floats: gcd(20,64)=4 -> 16 consecutive rows hit 16 distinct banks.
  __shared__ float ldsA[128][20];
  // Per-wave activation staging, stride 33 (odd) -> conflict-free transpose.
  __shared__ float ldsH[4][16][33];

  const int tid = threadIdx.x;
  const int idx = blockIdx.x * 128 + tid;

  // ----------------- Phase 1: quantum circuit (1 element / lane) -----------
  float cs[NQ], sn[NQ];
#pragma unroll
  for (int i = 0; i < NQ; ++i) {
    float h = x_q[idx * NQ + i] * 0.5f;
    __sincosf(h, &sn[i], &cs[i]);
  }

  float s[NSTATE];
#pragma unroll
  for (int i = 0; i < NSTATE; ++i) s[i] = 0.0f;
  s[0] = 1.0f;

  ry_all(s, cs, sn);

#pragma unroll
  for (int layer = 0; layer < 2; ++layer) {
    float cp[NQ], sp[NQ];
#pragma unroll
    for (int i = 0; i < NQ; ++i) {
      float h = q_params[layer * NQ + i] * 0.5f;   // uniform -> scalar loads
      __sincosf(h, &sp[i], &cp[i]);
    }
    ry_all(s, cp, sp);
    cnot_ring(s);
  }

  // <Z_0>: qubit 0 is amplitude-index bit 6.
  float pos = 0.0f, neg = 0.0f;
#pragma unroll
  for (int i = 0; i < 64; ++i)   pos += s[i] * s[i];
#pragma unroll
  for (int i = 64; i < 128; ++i) neg += s[i] * s[i];
  const float qexp = pos - neg;

  // Stage combined features [q, x_c[0..15], 0,0,0] as WMMA A-panel row.
  ldsA[tid][0] = qexp;
#pragma unroll
  for (int j = 0; j < 16; ++j) ldsA[tid][1 + j] = x_c[idx * 16 + j];
  ldsA[tid][17] = 0.0f;
  ldsA[tid][18] = 0.0f;
  ldsA[tid][19] = 0.0f;
  __syncthreads();

  // ----------------- Phase 2: MLP via V_WMMA_F32_16X16X4_F32 ---------------
  // Frag layouts (wave32):
  //   A (16x4):  lane l -> M = l%16;  a.x = A[M][2*half+0], a.y = A[M][2*half+1]
  //   B (4x16):  lane l -> N = l%16;  b.x = B[2*half+0][N], b.y = B[2*half+1][N]
  //   C/D:       vgpr r, lane l -> (M = r + 8*half, N = l%16)
  const int wave = tid >> 5;
  const int lane = tid & 31;
  const int half = lane >> 4;
  const int l16  = lane & 15;

  const float b3v = b3[0];
  const float w3l = w3[l16];   // w3 is (1,16)
  const float b2l = b2[l16];

#pragma unroll
  for (int t = 0; t < 2; ++t) {
    const int r0 = (wave * 2 + t) * 16;   // this wave's 16-row batch tile

    // ---- layer 1: (16x17) @ w1^T (17x32), K padded to 20, two N-tiles ----
#pragma unroll
    for (int nt = 0; nt < 2; ++nt) {
      const int n = nt * 16 + l16;
      v8f acc = {};
#pragma unroll
      for (int kc = 0; kc < 5; ++kc) {
        const int k0 = kc * 4;
        const int kk = k0 + half * 2;
        v2f a, b;
        a.x = ldsA[r0 + l16][kk + 0];
        a.y = ldsA[r0 + l16][kk + 1];
        // B[k][n] = w1[n][k], w1 row-major (32,17); zero beyond K=16.
        b.x = (kk     < 17) ? w1[n * 17 + kk    ] : 0.0f;
        b.y = (kk + 1 < 17) ? w1[n * 17 + kk + 1] : 0.0f;
        acc = __builtin_amdgcn_wmma_f32_16x16x4_f32(
            false, a, false, b, (short)0, acc, false, false);
      }
      const float bias = b1[n];
#pragma unroll
      for (int r = 0; r < 8; ++r) {
        float v = acc[r] + bias;
        ldsH[wave][r + half * 8][n] = v > 0.0f ? v : 0.0f;   // ReLU, stage h1
      }
    }
    // Cross-lane LDS RAW inside this wave: drain DS before reading back.
    asm volatile("s_wait_dscnt 0" ::: "memory");

    // ---- layer 2: (16x32) @ w2^T (32x16), 8 K-chunks ----
    v8f acc2 = {};
#pragma unroll
    for (int kc = 0; kc < 8; ++kc) {
      const int kk = kc * 4 + half * 2;
      v2f a, b;
      a.x = ldsH[wave][l16][kk + 0];
      a.y = ldsH[wave][l16][kk + 1];
      b.x = w2[l16 * 32 + kk + 0];   // B[k][n] = w2[n][k], w2 is (16,32)
      b.y = w2[l16 * 32 + kk + 1];
      acc2 = __builtin_amdgcn_wmma_f32_16x16x4_f32(
          false, a, false, b, (short)0, acc2, false, false);
    }

    // ---- layer 3: (16x16) @ w3^T (16x1) via shuffle reduction ----
#pragma unroll
    for (int r = 0; r < 8; ++r) {
      float v = acc2[r] + b2l;
      v = v > 0.0f ? v : 0.0f;          // ReLU
      float p = v * w3l;                // partial over N = l16
      p += __shfl_xor(p, 8, 16);
      p += __shfl_xor(p, 4, 16);
      p += __shfl_xor(p, 2, 16);
      p += __shfl_xor(p, 1, 16);
      if (l16 == 0)
        out[blockIdx.x * 128 + r0 + r + half * 8] = p + b3v;
    }
  }
}

extern "C" void kernel_launch(void* const* d_in, const int* in_sizes, int n_in,
                              void* d_out, int out_size, void* d_ws, size_t ws_size,
                              hipStream_t stream) {
  const float* x_q = (const float*)d_in[0];
  const float* x_c = (const float*)d_in[1];
  const float* qp  = (const float*)d_in[2];
  const float* w1  = (const float*)d_in[3];
  const float* b1  = (const float*)d_in[4];
  const float* w2  = (const float*)d_in[5];
  const float* b2  = (const float*)d_in[6];
  const float* w3  = (const float*)d_in[7];
  const float* b3  = (const float*)d_in[8];
  float* out = (float*)d_out;

  const int B = in_sizes[0] / NQ;        // 262144
  const int blocks = B / 128;            // 2048 blocks x 128 threads (4 waves)

  hipLaunchKernelGGL(hybrid_deepq_kernel, dim3(blocks), dim3(128), 0, stream,
                     x_q, x_c, qp, w1, b1, w2, b2, w3, b3, out);
}